// MBSTOI_88295937671397
// MI455X (gfx1250) — compile-verified
//
#include <hip/hip_runtime.h>
#include <math.h>

// Problem constants (from reference setup_inputs)
#define KCH   128
#define FFR   4000
#define NWIN  30
#define NSEG  (FFR - NWIN + 1)   // 3971
#define NBAND 15
#define NTAU  100
#define NGAM  40
#define TSPLIT 4                 // tau-chunks per segment (lanes cooperating)
#define TCHUNK (NTAU / TSPLIT)   // 25 taus per lane

typedef float v2f __attribute__((ext_vector_type(2)));
typedef float v8f __attribute__((ext_vector_type(8)));

// ---------------------------------------------------------------------------
// Kernel 1: per-frame band aggregates via WMMA f32 16x16x4.
//   agg[band(16)][prod(8)][frame(4000)] = sum_{k in band} prod(k, frame)
// One wave per (16-frame tile, product). D(16x16) = Mask(16x128) x Prod(128x16)
// accumulated in 32 K-steps of 4.
// Fragment layouts (CDNA5 ISA 7.12.2, f32 16x16x4):
//   A (16x4):  lane l -> M = l&15 ;  VGPR v -> K = v + 2*(l>>4)
//   B (4x16):  lane l -> N = l&15 ;  VGPR v -> K = v + 2*(l>>4)
//   C/D:       lane l -> N = l&15 ;  VGPR v -> M = v + 8*(l>>4)
// ---------------------------------------------------------------------------
__global__ __launch_bounds__(32) void band_agg_wmma(
    const float* __restrict__ xlre, const float* __restrict__ xlim,
    const float* __restrict__ xrre, const float* __restrict__ xrim,
    const float* __restrict__ ylre, const float* __restrict__ ylim,
    const float* __restrict__ yrre, const float* __restrict__ yrim,
    const long long* __restrict__ fids,
    float* __restrict__ agg)
{
  const int lane = threadIdx.x;
  const int f0   = blockIdx.x * 16;
  const int p    = blockIdx.y;

  // product p value at (k,f): P1*P2 + sgn*P3*P4
  const float *P1, *P2, *P3, *P4; float sgn = 1.0f;
  switch (p) {
    case 0: P1 = xlre; P2 = xlre; P3 = xlim; P4 = xlim; break;             // Lx
    case 1: P1 = xrre; P2 = xrre; P3 = xrim; P4 = xrim; break;             // Rx
    case 2: P1 = ylre; P2 = ylre; P3 = ylim; P4 = ylim; break;             // Ly
    case 3: P1 = yrre; P2 = yrre; P3 = yrim; P4 = yrim; break;             // Ry
    case 4: P1 = xrre; P2 = xlre; P3 = xrim; P4 = xlim; break;             // Re rhox
    case 5: P1 = xrre; P2 = xlim; P3 = xrim; P4 = xlre; sgn = -1.0f; break;// Im rhox
    case 6: P1 = yrre; P2 = ylre; P3 = yrim; P4 = ylim; break;             // Re rhoy
    default:P1 = yrre; P2 = ylim; P3 = yrim; P4 = ylre; sgn = -1.0f; break;// Im rhoy
  }

  const int m    = lane & 15;   // band row for A fragment
  const int half = lane >> 4;
  const int fcol = f0 + (lane & 15);

  int lo = 0, hi = 0;           // band [lo,hi) channel range (fids 1-based inclusive)
  if (m < NBAND) { lo = (int)fids[2*m] - 1; hi = (int)fids[2*m + 1]; }

  v8f acc = {0.f, 0.f, 0.f, 0.f, 0.f, 0.f, 0.f, 0.f};
  for (int k0 = 0; k0 < KCH; k0 += 4) {
    v2f a, b;
#pragma unroll
    for (int v = 0; v < 2; ++v) {
      const int k = k0 + v + 2 * half;
      a[v] = (k >= lo && k < hi) ? 1.0f : 0.0f;     // m>=NBAND has lo==hi==0 -> 0
      const int idx = k * FFR + fcol;
      b[v] = fmaf(P1[idx], P2[idx], sgn * P3[idx] * P4[idx]);
    }
    // 8-arg form: (neg_a, A, neg_b, B, c_mod, C, reuse_a, reuse_b)
    acc = __builtin_amdgcn_wmma_f32_16x16x4_f32(
        false, a, false, b, (short)0, acc, false, false);
  }

#pragma unroll
  for (int v = 0; v < 8; ++v) {
    const int mrow = v + 8 * half;
    agg[(mrow * 8 + p) * FFR + fcol] = acc[v];
  }
}

// ---------------------------------------------------------------------------
// Kernel 2: fused window stats + (tau,gamma) grid max.
// Grid: (ceil(S/64), NBAND), 256 threads. Four adjacent lanes share a segment,
// each scans 25 taus x 40 gammas, then a 2-step shuffle-max combines them.
// LDS holds the band-independent gamma table (a1,a2,b1,b2 per g) and the
// per-band tau table (cos wt, sin wt, cos 2wt, sin 2wt, Et=exp(-.5 c^2 sd^2)).
// Separability: epsdelexp[t,g] = Eg[g]*Et[t]; deltexp[t] = Et[t]^2.
// ---------------------------------------------------------------------------
__global__ __launch_bounds__(256) void segment_grid_max(
    const float* __restrict__ agg,
    const float* __restrict__ cf,   const float* __restrict__ taus,
    const float* __restrict__ gam,  const float* __restrict__ seps,
    const float* __restrict__ sdel,
    float* __restrict__ out)
{
  __shared__ float sG[NGAM * 4];
  __shared__ float sT[NTAU * 6];

  const int j   = blockIdx.y;
  const int tid = threadIdx.x;

  if (tid < NGAM) {
    const float L2_10  = 3.3219280948873623f;   // log2(10)
    const float LN10SQ = 5.3018981104783990f;   // ln(10)^2
    const float g  = gam[tid];
    const float se = seps[tid];
    const float epse = expf(2.0f * LN10SQ * se * se);
    const float Eg   = expf(0.5f * LN10SQ * se * se);
    sG[tid * 4 + 0] = exp2f( 2.0f * g * L2_10) * epse;  // g2p * epsexp
    sG[tid * 4 + 1] = exp2f(-2.0f * g * L2_10) * epse;  // g2m * epsexp
    sG[tid * 4 + 2] = exp2f( g * L2_10) * Eg;           // gp  * Eg
    sG[tid * 4 + 3] = exp2f(-g * L2_10) * Eg;           // gm  * Eg
  }
  if (tid >= 64 && tid < 64 + NTAU) {
    const int t = tid - 64;
    const float c  = cf[j];
    const float wt = c * taus[t];
    const float sd = sdel[t];
    sT[t * 6 + 0] = cosf(wt);
    sT[t * 6 + 1] = sinf(wt);
    sT[t * 6 + 2] = cosf(2.0f * wt);
    sT[t * 6 + 3] = sinf(2.0f * wt);
    sT[t * 6 + 4] = expf(-0.5f * c * c * sd * sd);      // Et
    sT[t * 6 + 5] = 0.0f;
  }
  __syncthreads();

  const int chunk = tid & (TSPLIT - 1);                 // which tau chunk
  const int s = blockIdx.x * (256 / TSPLIT) + (tid >> 2);
  if (s >= NSEG) return;                                // uniform across a quad

  const float* base = agg + (j * 8) * FFR + s;

  // pass 1: window means of the 8 aggregates
  float mean[8];
  {
    float acc[8] = {0.f, 0.f, 0.f, 0.f, 0.f, 0.f, 0.f, 0.f};
    for (int nn = 0; nn < NWIN; ++nn) {
#pragma unroll
      for (int q = 0; q < 8; ++q) acc[q] += base[q * FFR + nn];
    }
#pragma unroll
    for (int q = 0; q < 8; ++q) mean[q] = acc[q] * (1.0f / (float)NWIN);
  }

  // pass 2: 36 mean-removed dot products over the window
  float Sac=0,Sbd=0,Sad=0,Sbc=0;
  float Saqr=0,Scpr=0,Saqi=0,Scpi=0,Sbqr=0,Sdpr=0,Sbqi=0,Sdpi=0;
  float Sqrpr=0,Sqipi=0,Sqrpi=0,Sqipr=0;
  float Saa=0,Sbb=0,Sab=0,Sapr=0,Sapi=0,Sbpr=0,Sbpi=0,Sprpr=0,Spipi=0,Sprpi=0;
  float Scc=0,Sdd=0,Scd=0,Scqr=0,Scqi=0,Sdqr=0,Sdqi=0,Sqrqr=0,Sqiqi=0,Sqrqi=0;
  for (int nn = 0; nn < NWIN; ++nn) {
    const float a  = base[0*FFR+nn] - mean[0];  // Lx
    const float b  = base[1*FFR+nn] - mean[1];  // Rx
    const float c  = base[2*FFR+nn] - mean[2];  // Ly
    const float d  = base[3*FFR+nn] - mean[3];  // Ry
    const float pr = base[4*FFR+nn] - mean[4];  // Re rhox
    const float pi = base[5*FFR+nn] - mean[5];  // Im rhox
    const float qr = base[6*FFR+nn] - mean[6];  // Re rhoy
    const float qi = base[7*FFR+nn] - mean[7];  // Im rhoy
    Sac+=a*c; Sbd+=b*d; Sad+=a*d; Sbc+=b*c;
    Saqr+=a*qr; Scpr+=c*pr; Saqi+=a*qi; Scpi+=c*pi;
    Sbqr+=b*qr; Sdpr+=d*pr; Sbqi+=b*qi; Sdpi+=d*pi;
    Sqrpr+=qr*pr; Sqipi+=qi*pi; Sqrpi+=qr*pi; Sqipr+=qi*pr;
    Saa+=a*a; Sbb+=b*b; Sab+=a*b; Sapr+=a*pr; Sapi+=a*pi; Sbpr+=b*pr; Sbpi+=b*pi;
    Sprpr+=pr*pr; Spipi+=pi*pi; Sprpi+=pr*pi;
    Scc+=c*c; Sdd+=d*d; Scd+=c*d; Scqr+=c*qr; Scqi+=c*qi; Sdqr+=d*qr; Sdqi+=d*qi;
    Sqrqr+=qr*qr; Sqiqi+=qi*qi; Sqrqi+=qr*qi;
  }

  // per-pair grid-invariant scalars. exy = band_e(Lx,Ly,Rx,Ry, rhoy, rhox)
  const float Axy=Sac, Bxy=Sbd, Cxy=Sad+Sbc;
  const float a1xy=Saqr+Scpr, a2xy=Saqi+Scpi, b1xy=Sbqr+Sdpr, b2xy=Sbqi+Sdpi;
  const float t1xy=Sqrpr+Sqipi, qrexy=Sqrpr-Sqipi, qimxy=Sqrpi+Sqipr;

  const float Axx=Saa, Bxx=Sbb, Cxx=2.0f*Sab;
  const float a1xx=2.0f*Sapr, a2xx=2.0f*Sapi, b1xx=2.0f*Sbpr, b2xx=2.0f*Sbpi;
  const float t1xx=Sprpr+Spipi, qrexx=Sprpr-Spipi, qimxx=2.0f*Sprpi;

  const float Ayy=Scc, Byy=Sdd, Cyy=2.0f*Scd;
  const float a1yy=2.0f*Scqr, a2yy=2.0f*Scqi, b1yy=2.0f*Sdqr, b2yy=2.0f*Sdqi;
  const float t1yy=Sqrqr+Sqiqi, qreyy=Sqrqr-Sqiqi, qimyy=2.0f*Sqrqi;

  float best = -INFINITY;
  const int t0 = chunk * TCHUNK;
  for (int t = t0; t < t0 + TCHUNK; ++t) {
    const float ct = sT[t*6+0], st = sT[t*6+1];
    const float c2 = sT[t*6+2], s2 = sT[t*6+3];
    const float Et = sT[t*6+4];
    const float dx = Et * Et;                 // deltexp
    const float e2 = 2.0f * Et;

    const float uxy = e2 * fmaf(a1xy, ct, a2xy * st);
    const float vxy = e2 * fmaf(b1xy, ct, b2xy * st);
    const float base_xy = Cxy + 2.0f * fmaf(dx, fmaf(qrexy, c2, qimxy * s2), t1xy);
    const float uxx = e2 * fmaf(a1xx, ct, a2xx * st);
    const float vxx = e2 * fmaf(b1xx, ct, b2xx * st);
    const float base_xx = Cxx + 2.0f * fmaf(dx, fmaf(qrexx, c2, qimxx * s2), t1xx);
    const float uyy = e2 * fmaf(a1yy, ct, a2yy * st);
    const float vyy = e2 * fmaf(b1yy, ct, b2yy * st);
    const float base_yy = Cyy + 2.0f * fmaf(dx, fmaf(qreyy, c2, qimyy * s2), t1yy);

    for (int g = 0; g < NGAM; ++g) {
      const float ga1 = sG[4*g+0], ga2 = sG[4*g+1];
      const float gb1 = sG[4*g+2], gb2 = sG[4*g+3];
      const float Exy = fmaf(-uxy, gb1, fmaf(-vxy, gb2,
                        fmaf(Axy, ga1, fmaf(Bxy, ga2, base_xy))));
      const float Exx = fmaf(-uxx, gb1, fmaf(-vxx, gb2,
                        fmaf(Axx, ga1, fmaf(Bxx, ga2, base_xx))));
      const float Eyy = fmaf(-uyy, gb1, fmaf(-vyy, gb2,
                        fmaf(Ayy, ga1, fmaf(Byy, ga2, base_yy))));
      const float den = fmaxf(Exx * Eyy, 1e-10f);
      const float pv  = Exy * rsqrtf(den);
      best = fmaxf(best, pv);
    }
  }

  // combine the 4 tau-chunks held by adjacent lanes (wave32 cross-lane max)
  best = fmaxf(best, __shfl_xor(best, 1, 32));
  best = fmaxf(best, __shfl_xor(best, 2, 32));
  if (chunk == 0) out[j * NSEG + s] = best;
}

// ---------------------------------------------------------------------------
extern "C" void kernel_launch(void* const* d_in, const int* in_sizes, int n_in,
                              void* d_out, int out_size, void* d_ws, size_t ws_size,
                              hipStream_t stream) {
  (void)in_sizes; (void)n_in; (void)out_size; (void)ws_size;
  const float* xlre = (const float*)d_in[0];
  const float* xlim = (const float*)d_in[1];
  const float* xrre = (const float*)d_in[2];
  const float* xrim = (const float*)d_in[3];
  const float* ylre = (const float*)d_in[4];
  const float* ylim = (const float*)d_in[5];
  const float* yrre = (const float*)d_in[6];
  const float* yrim = (const float*)d_in[7];
  const float* cfp  = (const float*)d_in[8];
  const float* taus = (const float*)d_in[9];
  const float* gam  = (const float*)d_in[10];
  const float* seps = (const float*)d_in[11];
  const float* sdel = (const float*)d_in[12];
  const long long* fids = (const long long*)d_in[13];
  float* agg = (float*)d_ws;   // 16*8*4000 floats = 2 MB

  dim3 gB(FFR / 16, 8);
  band_agg_wmma<<<gB, 32, 0, stream>>>(xlre, xlim, xrre, xrim,
                                       ylre, ylim, yrre, yrim, fids, agg);

  const int segPerBlk = 256 / TSPLIT;  // 64
  dim3 gC((NSEG + segPerBlk - 1) / segPerBlk, NBAND);
  segment_grid_max<<<gC, 256, 0, stream>>>(agg, cfp, taus, gam, seps, sdel,
                                           (float*)d_out);
}